// ConsistencyBasedLaplacianBuilder_49340584296528
// MI455X (gfx1250) — compile-verified
//
#include <hip/hip_runtime.h>

// CDNA5 (gfx1250) wave32 WMMA types for V_WMMA_F32_16X16X4_F32:
//   A: 16x4 f32  -> 64 vals / 32 lanes = 2 VGPRs (v2f)
//   B: 4x16 f32  -> 2 VGPRs (v2f)
//   C/D: 16x16 f32 -> 8 VGPRs (v8f)
typedef __attribute__((ext_vector_type(2))) float v2f;
typedef __attribute__((ext_vector_type(8))) float v8f;

#define THREADS 256
#define MAX_BLOCKS 2048

// One edge per wave per iteration:
//   D = A1 * B1 + A2 * B2, A1 rows0-3 = Fvu, A2 rows0-3 = -Fuv, rows 4-15 zero
//   B1 = x[tgt] (4x16), B2 = x[src] (4x16)
// Rows 0-3 of D = diff; every other row is exactly zero (zero A rows), so
// sum of squares of the whole D accumulator == ||diff||_F^2, independent of
// the precise C/D register layout.
__global__ __launch_bounds__(THREADS) void sheaf_dirichlet_wmma(
    const float* __restrict__ x,          // [N,4,16]
    const float* __restrict__ maps,       // [E,4,4]
    const int*   __restrict__ edge_index, // [2,E]
    const int*   __restrict__ rev_idx,    // [E]
    float*       __restrict__ partials,   // [gridDim.x]
    int E)
{
  const int lane         = threadIdx.x & 31;
  const int waveInBlock  = threadIdx.x >> 5;
  const int wavesPerBlk  = THREADS / 32;
  const int waveId       = blockIdx.x * wavesPerBlk + waveInBlock;
  const int numWaves     = gridDim.x * wavesPerBlk;

  // ISA 16x4 f32 A layout: lanes 0-15 hold row m, K={0,1}; lanes 16-31 K={2,3}.
  const int m  = lane & 15;          // A row / B column index
  const int kb = (lane >> 4) << 1;   // K base: 0 or 2

  float acc = 0.0f;

  for (int e = waveId; e < E; e += numWaves) {
    const int srcN = edge_index[e];       // broadcast (same addr all lanes)
    const int tgtN = edge_index[E + e];
    const int rev  = rev_idx[e];

    // A operands: only rows 0-3 are real; others stay zero.
    v2f a1 = {0.f, 0.f};                  // Fvu  (maps[rev])
    v2f a2 = {0.f, 0.f};                  // -Fuv (maps[e])
    if (m < 4) {
      a1 = *(const v2f*)(maps + (size_t)rev * 16 + m * 4 + kb);   // 8B aligned
      v2f t = *(const v2f*)(maps + (size_t)e * 16 + m * 4 + kb);
      a2.x = -t.x;
      a2.y = -t.y;
    }

    // B operands: 4x16, VGPR0 = rows {kb}, VGPR1 = rows {kb+1} per half-wave.
    const float* xt = x + (size_t)tgtN * 64;
    const float* xs = x + (size_t)srcN * 64;
    v2f bt, bs;
    bt.x = xt[kb * 16 + m];
    bt.y = xt[(kb + 1) * 16 + m];
    bs.x = xs[kb * 16 + m];
    bs.y = xs[(kb + 1) * 16 + m];

    v8f c = {0.f, 0.f, 0.f, 0.f, 0.f, 0.f, 0.f, 0.f};
    // (neg_a, A, neg_b, B, c_mod, C, reuse_a, reuse_b)
    c = __builtin_amdgcn_wmma_f32_16x16x4_f32(false, a1, false, bt,
                                              (short)0, c, false, false);
    c = __builtin_amdgcn_wmma_f32_16x16x4_f32(false, a2, false, bs,
                                              (short)0, c, false, false);

    float s = c[0] * c[0];
    s = fmaf(c[1], c[1], s);
    s = fmaf(c[2], c[2], s);
    s = fmaf(c[3], c[3], s);
    s = fmaf(c[4], c[4], s);
    s = fmaf(c[5], c[5], s);
    s = fmaf(c[6], c[6], s);
    s = fmaf(c[7], c[7], s);
    acc += s;
  }

  // Deterministic in-block tree reduction.
  __shared__ float red[THREADS];
  red[threadIdx.x] = acc;
  __syncthreads();
  for (int off = THREADS / 2; off > 0; off >>= 1) {
    if (threadIdx.x < off) red[threadIdx.x] += red[threadIdx.x + off];
    __syncthreads();
  }
  if (threadIdx.x == 0) partials[blockIdx.x] = red[0];
}

// Single-block deterministic final reduction (double accumulation).
__global__ __launch_bounds__(THREADS) void reduce_partials(
    const float* __restrict__ partials, int n, float* __restrict__ out)
{
  __shared__ double red[THREADS];
  double a = 0.0;
  for (int i = threadIdx.x; i < n; i += THREADS) a += (double)partials[i];
  red[threadIdx.x] = a;
  __syncthreads();
  for (int off = THREADS / 2; off > 0; off >>= 1) {
    if (threadIdx.x < off) red[threadIdx.x] += red[threadIdx.x + off];
    __syncthreads();
  }
  if (threadIdx.x == 0) out[0] = (float)red[0];
}

extern "C" void kernel_launch(void* const* d_in, const int* in_sizes, int n_in,
                              void* d_out, int out_size, void* d_ws, size_t ws_size,
                              hipStream_t stream) {
  const float* x    = (const float*)d_in[0];   // [N,4,16] f32
  const float* maps = (const float*)d_in[1];   // [E,4,4]  f32
  const int*   ei   = (const int*)d_in[2];     // [2,E]    i32
  const int*   rev  = (const int*)d_in[3];     // [E]      i32
  const int    E    = in_sizes[3];

  size_t maxPartials = ws_size / sizeof(float);
  int blocks = (maxPartials < (size_t)MAX_BLOCKS) ? (int)maxPartials : MAX_BLOCKS;
  if (blocks < 1) blocks = 1;

  float* partials = (float*)d_ws;

  sheaf_dirichlet_wmma<<<blocks, THREADS, 0, stream>>>(x, maps, ei, rev,
                                                       partials, E);
  reduce_partials<<<1, THREADS, 0, stream>>>(partials, blocks, (float*)d_out);
}